// LocalGlobalRegistration_51316269253486
// MI455X (gfx1250) — compile-verified
//
#include <hip/hip_runtime.h>
#include <hip/hip_bf16.h>
#include <math.h>

// ---------------------------------------------------------------------------
// LocalGlobalRegistration for MI455X (gfx1250).
// Pipeline of small kernels; the 512x2048 inlier-count hotspot is mapped onto
// V_WMMA_F32_16X16X4_F32 via an exact rank-16 factorization of the squared
// residual (see analysis).  Everything fits in L2 (192MB); score_mat (34MB) is
// re-read with exp() recomputed instead of being materialized.
// ---------------------------------------------------------------------------

#define Bc   512
#define Nc   128
#define Cc   2048          // CORR_LIMIT
#define CAPc 262144        // max compacted positive correspondences (<= 196608 actual)
#define CONF_THRESH 0.05f
#define R2          0.01f  // ACCEPT_RADIUS^2
#define EPSc        1e-05f

typedef __attribute__((ext_vector_type(2))) float v2f;
typedef __attribute__((ext_vector_type(8))) float v8f;

// ---------------- shared helpers ----------------

__device__ __forceinline__ float block_reduce(float v, float* red) {
  int tid = threadIdx.x, n = blockDim.x;
  red[tid] = v; __syncthreads();
  for (int s = n >> 1; s > 0; s >>= 1) {
    if (tid < s) red[tid] += red[tid + s];
    __syncthreads();
  }
  float r = red[0]; __syncthreads();
  return r;
}

// masked score value (recomputed everywhere; bit-identical op order each time)
__device__ __forceinline__ float masked_score(const float* __restrict__ score,
                                              const unsigned char* __restrict__ rmask,
                                              const unsigned char* __restrict__ smask,
                                              const float* __restrict__ thr_row,
                                              const float* __restrict__ thr_col,
                                              int b, int r, int s) {
  float v = expf(score[(b * Nc + r) * Nc + s]);
  bool c = (v > CONF_THRESH) && (v >= thr_row[b * Nc + r]) && (v >= thr_col[b * Nc + s]) &&
           (rmask[b * Nc + r] != 0) && (smask[b * Nc + s] != 0);
  return c ? v : 0.0f;
}

__device__ __forceinline__ float det3c(const float M[3][3]) {
  return M[0][0] * (M[1][1] * M[2][2] - M[1][2] * M[2][1])
       - M[0][1] * (M[1][0] * M[2][2] - M[1][2] * M[2][0])
       + M[0][2] * (M[1][0] * M[2][1] - M[1][1] * M[2][0]);
}

// Kabsch from covariance H (H[c][d] = sum w*(src_c)(ref_d)), per reference:
// svd(H)=U S V^T ; R = V diag(1,1,sign(det(V U^T))) U^T ; t = rc - R sc.
__device__ void kabsch3(const float H[9], const float sc[3], const float rc[3], float T[16]) {
  float A[3][3], V[3][3];
  for (int i = 0; i < 3; ++i)
    for (int j = 0; j < 3; ++j) {
      float acc = 0.f;
      for (int k = 0; k < 3; ++k) acc += H[k * 3 + i] * H[k * 3 + j];  // H^T H
      A[i][j] = acc;
    }
  for (int i = 0; i < 3; ++i)
    for (int j = 0; j < 3; ++j) V[i][j] = (i == j) ? 1.f : 0.f;
  // cyclic Jacobi eigen-decomposition of A (symmetric)
  const int PQ[3][2] = {{0, 1}, {0, 2}, {1, 2}};
  for (int sweep = 0; sweep < 15; ++sweep) {
    for (int r = 0; r < 3; ++r) {
      int p = PQ[r][0], q = PQ[r][1];
      float apq = A[p][q];
      if (fabsf(apq) < 1e-24f) continue;
      float theta = (A[q][q] - A[p][p]) / (2.f * apq);
      float t = 1.f / (fabsf(theta) + sqrtf(theta * theta + 1.f));
      if (theta < 0.f) t = -t;
      float c = 1.f / sqrtf(t * t + 1.f), s = t * c;
      float app = A[p][p], aqq = A[q][q];
      A[p][p] = app - t * apq;
      A[q][q] = aqq + t * apq;
      A[p][q] = 0.f; A[q][p] = 0.f;
      int k = 3 - p - q;
      float akp = A[k][p], akq = A[k][q];
      A[k][p] = c * akp - s * akq; A[p][k] = A[k][p];
      A[k][q] = s * akp + c * akq; A[q][k] = A[k][q];
      for (int m = 0; m < 3; ++m) {
        float vp = V[m][p], vq = V[m][q];
        V[m][p] = c * vp - s * vq;
        V[m][q] = s * vp + c * vq;
      }
    }
  }
  float ev[3] = {A[0][0], A[1][1], A[2][2]};
  for (int i = 0; i < 2; ++i)
    for (int j = i + 1; j < 3; ++j)
      if (ev[j] > ev[i]) {
        float tmp = ev[i]; ev[i] = ev[j]; ev[j] = tmp;
        for (int m = 0; m < 3; ++m) { float tv = V[m][i]; V[m][i] = V[m][j]; V[m][j] = tv; }
      }
  // U columns: u_i = H v_i / sigma_i (+ Gram-Schmidt / degenerate fallbacks)
  float U[3][3];
  for (int i = 0; i < 3; ++i) {
    float w0 = H[0] * V[0][i] + H[1] * V[1][i] + H[2] * V[2][i];
    float w1 = H[3] * V[0][i] + H[4] * V[1][i] + H[5] * V[2][i];
    float w2 = H[6] * V[0][i] + H[7] * V[1][i] + H[8] * V[2][i];
    for (int j = 0; j < i; ++j) {
      float d = w0 * U[0][j] + w1 * U[1][j] + w2 * U[2][j];
      w0 -= d * U[0][j]; w1 -= d * U[1][j]; w2 -= d * U[2][j];
    }
    float n = sqrtf(w0 * w0 + w1 * w1 + w2 * w2);
    if (n > 1e-12f) { w0 /= n; w1 /= n; w2 /= n; }
    else if (i == 2) {
      w0 = U[1][0] * U[2][1] - U[2][0] * U[1][1];
      w1 = U[2][0] * U[0][1] - U[0][0] * U[2][1];
      w2 = U[0][0] * U[1][1] - U[1][0] * U[0][1];
    } else if (i == 1) {
      float ax = fabsf(U[0][0]), ay = fabsf(U[1][0]), az = fabsf(U[2][0]);
      float e0 = 0.f, e1 = 0.f, e2 = 0.f;
      if (ax <= ay && ax <= az) e0 = 1.f; else if (ay <= az) e1 = 1.f; else e2 = 1.f;
      float d = e0 * U[0][0] + e1 * U[1][0] + e2 * U[2][0];
      w0 = e0 - d * U[0][0]; w1 = e1 - d * U[1][0]; w2 = e2 - d * U[2][0];
      float nn = sqrtf(w0 * w0 + w1 * w1 + w2 * w2);
      w0 /= nn; w1 /= nn; w2 /= nn;
    } else { w0 = 1.f; w1 = 0.f; w2 = 0.f; }
    U[0][i] = w0; U[1][i] = w1; U[2][i] = w2;
  }
  float d = (det3c(U) * det3c(V)) < 0.f ? -1.f : 1.f;
  float R[3][3], tt[3];
  for (int i = 0; i < 3; ++i)
    for (int j = 0; j < 3; ++j)
      R[i][j] = V[i][0] * U[j][0] + V[i][1] * U[j][1] + d * V[i][2] * U[j][2];
  for (int i = 0; i < 3; ++i)
    tt[i] = rc[i] - (R[i][0] * sc[0] + R[i][1] * sc[1] + R[i][2] * sc[2]);
  T[0] = R[0][0]; T[1] = R[0][1]; T[2]  = R[0][2]; T[3]  = tt[0];
  T[4] = R[1][0]; T[5] = R[1][1]; T[6]  = R[1][2]; T[7]  = tt[1];
  T[8] = R[2][0]; T[9] = R[2][1]; T[10] = R[2][2]; T[11] = tt[2];
  T[12] = 0.f; T[13] = 0.f; T[14] = 0.f; T[15] = 1.f;
}

__device__ __forceinline__ float resid2_of(const float* T, const float* q, const float* s) {
  float a0 = T[0] * s[0] + T[1] * s[1] + T[2]  * s[2] + T[3];
  float a1 = T[4] * s[0] + T[5] * s[1] + T[6]  * s[2] + T[7];
  float a2 = T[8] * s[0] + T[9] * s[1] + T[10] * s[2] + T[11];
  float d0 = q[0] - a0, d1 = q[1] - a1, d2 = q[2] - a2;
  return d0 * d0 + d1 * d1 + d2 * d2;
}

// ---------------- kernels ----------------

// zero counters / histograms / selection buffers (ws is poisoned 0xAA)
__global__ void k_init(int* hist, int* counts, int* patch, int* ctr,
                       float* selScore, int* selIdx) {
  int i = blockIdx.x * blockDim.x + threadIdx.x;
  if (i < 2048) { hist[i] = 0; selScore[i] = 0.0f; selIdx[i] = 0; }
  if (i < Bc)   { counts[i] = 0; patch[i] = 0; }
  if (i < 8)    { ctr[i] = 0; }
}

// 3rd-largest of exp(score) along a row (last axis)
__global__ void k_row_thr(const float* __restrict__ score, float* __restrict__ thr_row) {
  int i = blockIdx.x * blockDim.x + threadIdx.x;
  if (i >= Bc * Nc) return;
  const float* row = score + i * Nc;
  float t1 = -1.f, t2 = -1.f, t3 = -1.f;
  for (int s = 0; s < Nc; ++s) {
    float v = expf(row[s]);
    if (v > t1)      { t3 = t2; t2 = t1; t1 = v; }
    else if (v > t2) { t3 = t2; t2 = v; }
    else if (v > t3) { t3 = v; }
  }
  thr_row[i] = t3;
}

// 3rd-largest of exp(score) along a column
__global__ void k_col_thr(const float* __restrict__ score, float* __restrict__ thr_col) {
  int i = blockIdx.x * blockDim.x + threadIdx.x;
  if (i >= Bc * Nc) return;
  int b = i / Nc, s = i % Nc;
  float t1 = -1.f, t2 = -1.f, t3 = -1.f;
  for (int r = 0; r < Nc; ++r) {
    float v = expf(score[(b * Nc + r) * Nc + s]);
    if (v > t1)      { t3 = t2; t2 = t1; t1 = v; }
    else if (v > t2) { t3 = t2; t2 = v; }
    else if (v > t3) { t3 = v; }
  }
  thr_col[i] = t3;
}

__global__ void k_rowsum_patch(const float* score, const unsigned char* rmask,
                               const unsigned char* smask, const float* thr_row,
                               const float* thr_col, float* rowsum, int* patch) {
  int i = blockIdx.x * blockDim.x + threadIdx.x;
  if (i >= Bc * Nc) return;
  int b = i / Nc, r = i % Nc;
  float acc = 0.f; int cnt = 0;
  for (int s = 0; s < Nc; ++s) {
    float v = masked_score(score, rmask, smask, thr_row, thr_col, b, r, s);
    acc += v; cnt += (v > 0.f);
  }
  rowsum[i] = acc;
  atomicAdd(&patch[b], cnt);
}

__global__ void k_colsum(const float* score, const unsigned char* rmask,
                         const unsigned char* smask, const float* thr_row,
                         const float* thr_col, float* colsum) {
  int i = blockIdx.x * blockDim.x + threadIdx.x;
  if (i >= Bc * Nc) return;
  int b = i / Nc, s = i % Nc;
  float acc = 0.f;
  for (int r = 0; r < Nc; ++r)
    acc += masked_score(score, rmask, smask, thr_row, thr_col, b, r, s);
  colsum[i] = acc;
}

__global__ void k_ssum(const float* rowsum, float* Ssum) {
  int b = blockIdx.x * blockDim.x + threadIdx.x;
  if (b >= Bc) return;
  float acc = 0.f;
  for (int r = 0; r < Nc; ++r) acc += rowsum[b * Nc + r];
  Ssum[b] = acc;
}

// sc = sum_m colsum_wn[m]*src[m,:] ; rc = sum_n rowsum_wn[n]*ref[n,:]
__global__ void k_centroid(const float* rowsum, const float* colsum, const float* Ssum,
                           const float* __restrict__ src, const float* __restrict__ ref,
                           float* sc, float* rc) {
  int b = blockIdx.x * blockDim.x + threadIdx.x;
  if (b >= Bc) return;
  float inv = 1.f / (Ssum[b] + EPSc);
  float s0 = 0.f, s1 = 0.f, s2 = 0.f, r0 = 0.f, r1 = 0.f, r2 = 0.f;
  for (int m = 0; m < Nc; ++m) {
    float wc = colsum[b * Nc + m] * inv;
    float wr = rowsum[b * Nc + m] * inv;
    const float* sp = src + (b * Nc + m) * 3;
    const float* rp = ref + (b * Nc + m) * 3;
    s0 += wc * sp[0]; s1 += wc * sp[1]; s2 += wc * sp[2];
    r0 += wr * rp[0]; r1 += wr * rp[1]; r2 += wr * rp[2];
  }
  sc[b * 3 + 0] = s0; sc[b * 3 + 1] = s1; sc[b * 3 + 2] = s2;
  rc[b * 3 + 0] = r0; rc[b * 3 + 1] = r1; rc[b * 3 + 2] = r2;
}

// Hb[c][d] = sum_{n,m} wn[n,m]*(src[m,c]-sc_c)*(ref[n,d]-rc_d)
__global__ __launch_bounds__(Nc) void k_hb(const float* score, const unsigned char* rmask,
                                           const unsigned char* smask, const float* thr_row,
                                           const float* thr_col, const float* Ssum,
                                           const float* sc, const float* rc,
                                           const float* __restrict__ src,
                                           const float* __restrict__ ref, float* Hb) {
  __shared__ float red[Nc];
  int b = blockIdx.x, n = threadIdx.x;
  float inv = 1.f / (Ssum[b] + EPSc);
  float sc0 = sc[b * 3 + 0], sc1 = sc[b * 3 + 1], sc2 = sc[b * 3 + 2];
  float a0 = 0.f, a1 = 0.f, a2 = 0.f;
  for (int m = 0; m < Nc; ++m) {
    float v = masked_score(score, rmask, smask, thr_row, thr_col, b, n, m);
    if (v > 0.f) {
      float w = v * inv;
      const float* sp = src + (b * Nc + m) * 3;
      a0 += w * (sp[0] - sc0); a1 += w * (sp[1] - sc1); a2 += w * (sp[2] - sc2);
    }
  }
  const float* rp = ref + (b * Nc + n) * 3;
  float rd0 = rp[0] - rc[b * 3 + 0], rd1 = rp[1] - rc[b * 3 + 1], rd2 = rp[2] - rc[b * 3 + 2];
  float h[9] = {a0 * rd0, a0 * rd1, a0 * rd2,
                a1 * rd0, a1 * rd1, a1 * rd2,
                a2 * rd0, a2 * rd1, a2 * rd2};
  for (int j = 0; j < 9; ++j) {
    float t = block_reduce(h[j], red);
    if (n == 0) Hb[b * 9 + j] = t;
  }
}

// per-batch SVD -> local transform + psi features for the WMMA residual GEMM
__global__ void k_svd_psi(const float* Hb, const float* sc, const float* rc,
                          float* TRt, float* psi) {
  int b = blockIdx.x * blockDim.x + threadIdx.x;
  if (b >= Bc) return;
  float H[9], scv[3], rcv[3], T[16];
  for (int j = 0; j < 9; ++j) H[j] = Hb[b * 9 + j];
  for (int j = 0; j < 3; ++j) { scv[j] = sc[b * 3 + j]; rcv[j] = rc[b * 3 + j]; }
  kabsch3(H, scv, rcv, T);
  float R[3][3] = {{T[0], T[1], T[2]}, {T[4], T[5], T[6]}, {T[8], T[9], T[10]}};
  float t0 = T[3], t1 = T[7], t2 = T[11];
  float* o = TRt + b * 12;
  o[0] = R[0][0]; o[1] = R[0][1]; o[2] = R[0][2];
  o[3] = R[1][0]; o[4] = R[1][1]; o[5] = R[1][2];
  o[6] = R[2][0]; o[7] = R[2][1]; o[8] = R[2][2];
  o[9] = t0; o[10] = t1; o[11] = t2;
  // psi: resid^2(p,c) = (|q|^2+|s|^2) + phi(c).psi(p)
  float* p = psi + b * 16;
  p[0] = t0 * t0 + t1 * t1 + t2 * t2;
  p[1] = -2.f * t0; p[2] = -2.f * t1; p[3] = -2.f * t2;
  for (int i = 0; i < 3; ++i)
    for (int j = 0; j < 3; ++j) p[4 + 3 * i + j] = -2.f * R[i][j];
  for (int j = 0; j < 3; ++j)
    p[13 + j] = 2.f * (R[0][j] * t0 + R[1][j] * t1 + R[2][j] * t2);
}

// compact all positive masked scores (value, flat index)
__global__ void k_compact(const float* score, const unsigned char* rmask,
                          const unsigned char* smask, const float* thr_row,
                          const float* thr_col, float* cvals, int* cidx, int* ctr) {
  int i = blockIdx.x * blockDim.x + threadIdx.x;
  if (i >= Bc * Nc * Nc) return;
  int b = i / (Nc * Nc), rem = i % (Nc * Nc);
  float v = masked_score(score, rmask, smask, thr_row, thr_col, b, rem / Nc, rem % Nc);
  if (v > 0.f) {
    int pos = atomicAdd(&ctr[0], 1);
    if (pos < CAPc) { cvals[pos] = v; cidx[pos] = i; }
  }
}

__global__ void k_hist(const float* cvals, const int* ctr, int* hist) {
  int i = blockIdx.x * blockDim.x + threadIdx.x;
  int M = min(ctr[0], CAPc);
  if (i >= M) return;
  unsigned bin = __float_as_uint(cvals[i]) >> 21;   // positive floats: monotonic
  atomicAdd(&hist[bin], 1);
}

__global__ void k_sel_thresh(const int* hist, int* ctr) {
  int M = min(ctr[0], CAPc);
  int Tb = -1, hi = 0;
  if (M > Cc) {
    int cum = 0;
    for (int bin = 2047; bin >= 0; --bin) {
      int h = hist[bin];
      if (cum + h >= Cc) { Tb = bin; hi = cum; break; }
      cum += h;
    }
  }
  ctr[3] = Tb; ctr[4] = hi;
}

__global__ void k_sel_compact(const float* cvals, const int* cidx, const int* ctr,
                              float* selScore, int* selIdx, int* ctrw) {
  int i = blockIdx.x * blockDim.x + threadIdx.x;
  int M = min(ctr[0], CAPc);
  if (i >= M) return;
  float v = cvals[i];
  int bin = (int)(__float_as_uint(v) >> 21);
  int Tb = ctr[3], hi = ctr[4];
  if (bin > Tb) {
    int p = atomicAdd(&ctrw[1], 1);
    if (p < Cc) { selScore[p] = v; selIdx[p] = cidx[i]; }
  } else if (bin == Tb) {
    int e = atomicAdd(&ctrw[2], 1);
    int p = hi + e;
    if (p < Cc) { selScore[p] = v; selIdx[p] = cidx[i]; }
  }
}

// build phi features + per-correspondence threshold; stash q/s for refinement
__global__ void k_phi(const float* selScore, const int* selIdx,
                      const float* __restrict__ ref, const float* __restrict__ src,
                      float* phi, float* thrEff, float* selQ, float* selS) {
  int c = blockIdx.x * blockDim.x + threadIdx.x;
  if (c >= Cc) return;
  int idx = selIdx[c];
  int b = idx / (Nc * Nc), rem = idx % (Nc * Nc);
  int r = rem / Nc, s = rem % Nc;
  const float* qp = ref + (b * Nc + r) * 3;
  const float* sp = src + (b * Nc + s) * 3;
  float q0 = qp[0], q1 = qp[1], q2 = qp[2];
  float s0 = sp[0], s1 = sp[1], s2 = sp[2];
  float* f = phi + c * 16;
  f[0] = 1.f;
  f[1] = q0; f[2] = q1; f[3] = q2;
  f[4]  = q0 * s0; f[5]  = q0 * s1; f[6]  = q0 * s2;
  f[7]  = q1 * s0; f[8]  = q1 * s1; f[9]  = q1 * s2;
  f[10] = q2 * s0; f[11] = q2 * s1; f[12] = q2 * s2;
  f[13] = s0; f[14] = s1; f[15] = s2;
  float a = q0 * q0 + q1 * q1 + q2 * q2 + s0 * s0 + s1 * s1 + s2 * s2;
  thrEff[c] = (selScore[c] > 0.f) ? (R2 - a) : -1e30f;   // invalid -> never inlier
  selQ[c * 3 + 0] = q0; selQ[c * 3 + 1] = q1; selQ[c * 3 + 2] = q2;
  selS[c * 3 + 0] = s0; selS[c * 3 + 1] = s1; selS[c * 3 + 2] = s2;
}

// 2048x512 inlier-count GEMM: D = Phi(2048x16) * Psi^T(16x512) via
// V_WMMA_F32_16X16X4_F32, one wave per 16x16 tile, K=16 as 4 chained WMMAs.
__global__ __launch_bounds__(32) void k_count_wmma(const float* __restrict__ phi,
                                                   const float* __restrict__ psi,
                                                   const float* __restrict__ thrEff,
                                                   int* __restrict__ counts) {
  int c0 = blockIdx.x * 16;        // correspondence tile
  int p0 = blockIdx.y * 16;        // transform tile
  int lane = threadIdx.x;          // wave32
  int l = lane & 15, half = lane >> 4;
  v8f acc = {};
#pragma unroll
  for (int k = 0; k < 4; ++k) {
    int fo = 4 * k + 2 * half;     // A: lanes0-15 hold K={4k,4k+1}, lanes16-31 K={4k+2,4k+3}
    v2f a, bm;
    a.x  = phi[(c0 + l) * 16 + fo];
    a.y  = phi[(c0 + l) * 16 + fo + 1];
    bm.x = psi[(p0 + l) * 16 + fo];      // B[K][n] = Psi[n][K], mirrored layout
    bm.y = psi[(p0 + l) * 16 + fo + 1];
    acc = __builtin_amdgcn_wmma_f32_16x16x4_f32(false, a, false, bm,
                                                (short)0, acc, false, false);
  }
  // D layout: VGPR i -> M = i (lanes 0-15) or 8+i (lanes 16-31), N = lane&15
  int cnt = 0;
#pragma unroll
  for (int i = 0; i < 8; ++i) {
    int m = c0 + i + 8 * half;
    cnt += (acc[i] < thrEff[m]) ? 1 : 0;
  }
  cnt += __shfl_xor(cnt, 16, 32);
  if (half == 0) atomicAdd(&counts[p0 + l], cnt);
}

// counts gating + argmax (first max wins, matching jnp.argmax)
__global__ __launch_bounds__(Bc) void k_best(const int* counts, const int* patch, int* ctr) {
  __shared__ int bv[Bc], bi[Bc];
  int p = threadIdx.x;
  bv[p] = (patch[p] >= 3) ? counts[p] : -1;
  bi[p] = p;
  __syncthreads();
  for (int s = Bc / 2; s > 0; s >>= 1) {
    if (p < s) {
      if (bv[p + s] > bv[p] || (bv[p + s] == bv[p] && bi[p + s] < bi[p])) {
        bv[p] = bv[p + s]; bi[p] = bi[p + s];
      }
    }
    __syncthreads();
  }
  if (p == 0) ctr[5] = bi[0];
}

// 5 rounds of weighted Procrustes over the 2048 selected correspondences
__global__ __launch_bounds__(256) void k_refine(const float* selScore, const float* selQ,
                                                const float* selS, const float* TRt,
                                                const int* ctr, float* outT) {
  __shared__ float red[256];
  __shared__ float curW[Cc];
  __shared__ float Tsh[16];
  __shared__ float stats[8];   // sc(3), rc(3), invW
  int tid = threadIdx.x;
  if (tid == 0) {
    const float* o = TRt + ctr[5] * 12;
    Tsh[0] = o[0]; Tsh[1] = o[1]; Tsh[2]  = o[2]; Tsh[3]  = o[9];
    Tsh[4] = o[3]; Tsh[5] = o[4]; Tsh[6]  = o[5]; Tsh[7]  = o[10];
    Tsh[8] = o[6]; Tsh[9] = o[7]; Tsh[10] = o[8]; Tsh[11] = o[11];
    Tsh[12] = 0.f; Tsh[13] = 0.f; Tsh[14] = 0.f; Tsh[15] = 1.f;
  }
  __syncthreads();
  // initial weights from best local transform (invalid rows have score 0)
  for (int j = tid; j < Cc; j += 256) {
    float w = selScore[j] *
              ((resid2_of(Tsh, selQ + j * 3, selS + j * 3) < R2) ? 1.f : 0.f);
    curW[j] = w;
  }
  __syncthreads();
  for (int it = 0; it < 5; ++it) {
    float sw = 0.f, ss0 = 0.f, ss1 = 0.f, ss2 = 0.f, sq0 = 0.f, sq1 = 0.f, sq2 = 0.f;
    for (int j = tid; j < Cc; j += 256) {
      float w = curW[j];
      const float* s = selS + j * 3;
      const float* q = selQ + j * 3;
      sw += w;
      ss0 += w * s[0]; ss1 += w * s[1]; ss2 += w * s[2];
      sq0 += w * q[0]; sq1 += w * q[1]; sq2 += w * q[2];
    }
    sw  = block_reduce(sw, red);
    ss0 = block_reduce(ss0, red); ss1 = block_reduce(ss1, red); ss2 = block_reduce(ss2, red);
    sq0 = block_reduce(sq0, red); sq1 = block_reduce(sq1, red); sq2 = block_reduce(sq2, red);
    if (tid == 0) {
      float inv = 1.f / (sw + EPSc);
      stats[0] = ss0 * inv; stats[1] = ss1 * inv; stats[2] = ss2 * inv;
      stats[3] = sq0 * inv; stats[4] = sq1 * inv; stats[5] = sq2 * inv;
      stats[6] = inv;
    }
    __syncthreads();
    float sc0 = stats[0], sc1 = stats[1], sc2 = stats[2];
    float rc0 = stats[3], rc1 = stats[4], rc2 = stats[5];
    float inv = stats[6];
    float h[9] = {0.f, 0.f, 0.f, 0.f, 0.f, 0.f, 0.f, 0.f, 0.f};
    for (int j = tid; j < Cc; j += 256) {
      float wn = curW[j] * inv;
      const float* s = selS + j * 3;
      const float* q = selQ + j * 3;
      float d0 = s[0] - sc0, d1 = s[1] - sc1, d2 = s[2] - sc2;
      float e0 = q[0] - rc0, e1 = q[1] - rc1, e2 = q[2] - rc2;
      h[0] += wn * d0 * e0; h[1] += wn * d0 * e1; h[2] += wn * d0 * e2;
      h[3] += wn * d1 * e0; h[4] += wn * d1 * e1; h[5] += wn * d1 * e2;
      h[6] += wn * d2 * e0; h[7] += wn * d2 * e1; h[8] += wn * d2 * e2;
    }
    float Hm[9];
    for (int j = 0; j < 9; ++j) Hm[j] = block_reduce(h[j], red);
    if (tid == 0) {
      float scv[3] = {sc0, sc1, sc2}, rcv[3] = {rc0, rc1, rc2};
      kabsch3(Hm, scv, rcv, Tsh);
    }
    __syncthreads();
    if (it < 4) {
      for (int j = tid; j < Cc; j += 256) {
        curW[j] = selScore[j] *
                  ((resid2_of(Tsh, selQ + j * 3, selS + j * 3) < R2) ? 1.f : 0.f);
      }
      __syncthreads();
    }
  }
  if (tid < 16) outT[tid] = Tsh[tid];
}

// ---------------- host launcher ----------------

extern "C" void kernel_launch(void* const* d_in, const int* in_sizes, int n_in,
                              void* d_out, int out_size, void* d_ws, size_t ws_size,
                              hipStream_t stream) {
  const float* score = (const float*)d_in[0];            // (512,128,128) f32
  const float* ref   = (const float*)d_in[1];            // (512,128,3)   f32
  const float* src   = (const float*)d_in[2];            // (512,128,3)   f32
  const unsigned char* rmask = (const unsigned char*)d_in[3];  // (512,128) bool (1B)
  const unsigned char* smask = (const unsigned char*)d_in[4];  // (512,128) bool (1B)
  float* outT = (float*)d_out;                           // (4,4) f32

  // workspace carve-out (all 256B-aligned; total ~3.5 MB)
  char* w = (char*)d_ws;
  auto carve = [&](size_t bytes) {
    void* p = (void*)w;
    w += (bytes + 255) & ~(size_t)255;
    return p;
  };
  float* thr_row = (float*)carve(Bc * Nc * 4);
  float* thr_col = (float*)carve(Bc * Nc * 4);
  float* rowsum  = (float*)carve(Bc * Nc * 4);
  float* colsum  = (float*)carve(Bc * Nc * 4);
  float* Ssum    = (float*)carve(Bc * 4);
  float* sc      = (float*)carve(Bc * 3 * 4);
  float* rc      = (float*)carve(Bc * 3 * 4);
  float* Hb      = (float*)carve(Bc * 9 * 4);
  float* TRt     = (float*)carve(Bc * 12 * 4);
  float* psi     = (float*)carve(Bc * 16 * 4);
  float* phi     = (float*)carve(Cc * 16 * 4);
  float* thrEff  = (float*)carve(Cc * 4);
  float* selScore= (float*)carve(Cc * 4);
  float* selQ    = (float*)carve(Cc * 3 * 4);
  float* selS    = (float*)carve(Cc * 3 * 4);
  float* cvals   = (float*)carve(CAPc * 4);
  int*   cidx    = (int*)  carve(CAPc * 4);
  int*   selIdx  = (int*)  carve(Cc * 4);
  int*   hist    = (int*)  carve(2048 * 4);
  int*   counts  = (int*)  carve(Bc * 4);
  int*   patch   = (int*)  carve(Bc * 4);
  int*   ctr     = (int*)  carve(8 * 4);

  const int TB = 256;
  k_init<<<(2048 + TB - 1) / TB, TB, 0, stream>>>(hist, counts, patch, ctr, selScore, selIdx);
  k_row_thr<<<(Bc * Nc + TB - 1) / TB, TB, 0, stream>>>(score, thr_row);
  k_col_thr<<<(Bc * Nc + TB - 1) / TB, TB, 0, stream>>>(score, thr_col);
  k_rowsum_patch<<<(Bc * Nc + TB - 1) / TB, TB, 0, stream>>>(score, rmask, smask, thr_row,
                                                             thr_col, rowsum, patch);
  k_colsum<<<(Bc * Nc + TB - 1) / TB, TB, 0, stream>>>(score, rmask, smask, thr_row,
                                                       thr_col, colsum);
  k_ssum<<<(Bc + TB - 1) / TB, TB, 0, stream>>>(rowsum, Ssum);
  k_centroid<<<(Bc + TB - 1) / TB, TB, 0, stream>>>(rowsum, colsum, Ssum, src, ref, sc, rc);
  k_hb<<<Bc, Nc, 0, stream>>>(score, rmask, smask, thr_row, thr_col, Ssum, sc, rc, src, ref, Hb);
  k_svd_psi<<<(Bc + TB - 1) / TB, TB, 0, stream>>>(Hb, sc, rc, TRt, psi);
  k_compact<<<(Bc * Nc * Nc + TB - 1) / TB, TB, 0, stream>>>(score, rmask, smask, thr_row,
                                                             thr_col, cvals, cidx, ctr);
  k_hist<<<(CAPc + TB - 1) / TB, TB, 0, stream>>>(cvals, ctr, hist);
  k_sel_thresh<<<1, 1, 0, stream>>>(hist, ctr);
  k_sel_compact<<<(CAPc + TB - 1) / TB, TB, 0, stream>>>(cvals, cidx, ctr, selScore, selIdx, ctr);
  k_phi<<<(Cc + TB - 1) / TB, TB, 0, stream>>>(selScore, selIdx, ref, src, phi, thrEff, selQ, selS);
  dim3 cg(Cc / 16, Bc / 16);            // 128 x 32 tiles, one wave each
  k_count_wmma<<<cg, 32, 0, stream>>>(phi, psi, thrEff, counts);
  k_best<<<1, Bc, 0, stream>>>(counts, patch, ctr);
  k_refine<<<1, 256, 0, stream>>>(selScore, selQ, selS, TRt, ctr, outT);
}